// PolyToImage2D_12240656794193
// MI455X (gfx1250) — compile-verified
//
#include <hip/hip_runtime.h>
#include <hip/hip_bf16.h>
#include <math.h>

// ---------------------------------------------------------------------------
// TPS warp + bilinear sample for B=8, N=32 control pts, C=3, H=W=512.
// Stage 1: per-batch 35x35 Gaussian-elimination solve (tiny) -> params in d_ws
// Stage 2: fused TPS-apply + grid-sample.
//   r^2 against all 32 sources = rank-4 GEMM -> V_WMMA_F32_16X16X4_F32.
//   Two 16-pixel tiles per iteration (4 WMMAs); u = 0.5 r^2 ln(r^2+eps) uses
//   raw v_log_f32 (log2) with 0.5*ln2 folded into the TPS weights; the
//   cross-half source reduction is one shfl_xor(16) per accumulator; lane
//   halves then split the two tiles for full-EXEC bilinear sampling.
// ---------------------------------------------------------------------------

typedef __attribute__((ext_vector_type(2))) float v2f;
typedef __attribute__((ext_vector_type(8))) float v8f;

#define HD 512
#define WD 512
#define NPTS 32
#define NSYS 35   // 32 + 3 affine rows
#define EPSF 1e-7f
#define HALF_LN2 0.34657359027997264f   // 0.5 * ln(2)

// ---------------- Stage 1: TPS fit (one block per batch) -------------------
__global__ void tps_fit_kernel(const float* __restrict__ x,     // (8,32,2)
                               const float* __restrict__ poly,  // (1,32,2)
                               float* __restrict__ params)      // (8,35,2)
{
    __shared__ float Laug[NSYS][38];  // 35 cols of L + 2 RHS cols (pad to 38)
    const int b = blockIdx.x;
    const int t = threadIdx.x;        // blockDim.x == 64
    const float* xb = x + b * NPTS * 2;

    // Build augmented system [L | Y]
    for (int i = t; i < NSYS; i += blockDim.x) {
        if (i < NPTS) {
            const float xi = xb[2*i], yi = xb[2*i+1];
            for (int j = 0; j < NPTS; ++j) {
                const float dx = xi - xb[2*j], dy = yi - xb[2*j+1];
                const float r2 = dx*dx + dy*dy;
                Laug[i][j] = 0.5f * r2 * __logf(r2 + EPSF);
            }
            Laug[i][32] = 1.0f; Laug[i][33] = xi; Laug[i][34] = yi;
            Laug[i][35] = poly[2*i]; Laug[i][36] = poly[2*i+1];
        } else {
            const int k = i - NPTS;   // 0: ones row, 1: x row, 2: y row (P^T)
            for (int j = 0; j < NPTS; ++j)
                Laug[i][j] = (k == 0) ? 1.0f : xb[2*j + (k-1)];
            Laug[i][32] = 0.0f; Laug[i][33] = 0.0f; Laug[i][34] = 0.0f;
            Laug[i][35] = 0.0f; Laug[i][36] = 0.0f;
        }
    }
    __syncthreads();

    // Gaussian elimination with partial pivoting
    for (int k = 0; k < NSYS; ++k) {
        if (t == 0) {
            int piv = k; float mx = fabsf(Laug[k][k]);
            for (int i = k + 1; i < NSYS; ++i) {
                const float v = fabsf(Laug[i][k]);
                if (v > mx) { mx = v; piv = i; }
            }
            if (piv != k)
                for (int j = k; j < 37; ++j) {
                    const float tmp = Laug[k][j];
                    Laug[k][j] = Laug[piv][j];
                    Laug[piv][j] = tmp;
                }
        }
        __syncthreads();
        const float pivot = Laug[k][k];
        for (int i = t; i < NSYS; i += blockDim.x) {
            if (i > k) {
                const float f = Laug[i][k] / pivot;
                for (int j = k; j < 37; ++j)
                    Laug[i][j] = fmaf(-f, Laug[k][j], Laug[i][j]);
            }
        }
        __syncthreads();
    }

    // Back substitution: thread 0 -> RHS 0, thread 1 -> RHS 1
    if (t < 2) {
        float sol[NSYS];
        for (int k = NSYS - 1; k >= 0; --k) {
            float s = Laug[k][35 + t];
            for (int j = k + 1; j < NSYS; ++j) s -= Laug[k][j] * sol[j];
            sol[k] = s / Laug[k][k];
        }
        for (int k = 0; k < NSYS; ++k)
            params[b * (NSYS * 2) + 2 * k + t] = sol[k];
    }
}

// ---------------- Stage 2: fused TPS apply + bilinear sample ---------------
// r^2(j,p) = [-2sx_j, -2sy_j, 1, |s_j|^2] . [gx_p, gy_p, |g_p|^2, 1]
// A frag (16x4 f32): lane L -> M=L&15; low half K=0,1; high half K=2,3.
// B frag (4x16 f32): assumed row-striped: VGPR0={K0 lanes0-15, K2 lanes16-31},
//                    VGPR1={K1,K3}.
// D frag: lane L, reg v -> source M = v + 8*(L>=16), pixel N = L&15.
__global__ void __launch_bounds__(256)
tps_warp_sample_kernel(const float* __restrict__ x,       // (8,32,2)
                       const float* __restrict__ image,   // (1,3,512,512)
                       const float* __restrict__ params,  // (8,35,2)
                       float* __restrict__ out)           // (8,3,512,512)
{
    const int blocksPerBatch = 64;
    const int b      = blockIdx.x / blocksPerBatch;
    const int blkInB = blockIdx.x % blocksPerBatch;
    const int wave   = threadIdx.x >> 5;
    const int lane   = threadIdx.x & 31;
    const int m      = lane & 15;
    const bool lo    = lane < 16;
    const int hi8    = lo ? 0 : 8;

    // Per-lane A-fragment constants (sources for this batch)
    const float* xb = x + b * NPTS * 2;
    const float sx0 = xb[2*m],        sy0 = xb[2*m+1];
    const float sx1 = xb[2*(m+16)],   sy1 = xb[2*(m+16)+1];
    v2f a0, a1;
    a0.x = lo ? -2.0f * sx0 : 1.0f;
    a0.y = lo ? -2.0f * sy0 : fmaf(sx0, sx0, sy0 * sy0);
    a1.x = lo ? -2.0f * sx1 : 1.0f;
    a1.y = lo ? -2.0f * sy1 : fmaf(sx1, sx1, sy1 * sy1);

    // Per-lane TPS weights matching the D-fragment's source rows.
    // Pre-scaled by 0.5*ln2 so u*w = (r2*log2(r2+eps)) * w'.
    const float* prm = params + b * (NSYS * 2);
    float wgx[16], wgy[16];
    #pragma unroll
    for (int tt = 0; tt < 2; ++tt)
        #pragma unroll
        for (int v = 0; v < 8; ++v) {
            const int j = 16 * tt + hi8 + v;
            wgx[8 * tt + v] = prm[2 * j]     * HALF_LN2;
            wgy[8 * tt + v] = prm[2 * j + 1] * HALF_LN2;
        }
    const float A00 = prm[64], A01 = prm[65];   // affine const
    const float A10 = prm[66], A11 = prm[67];   // * gx
    const float A20 = prm[68], A21 = prm[69];   // * gy

    const int wavesPerBatch = blocksPerBatch * 8;            // 512
    const int waveId  = blkInB * 8 + wave;                   // 0..511
    const int nTiles  = (HD * WD) / 16;                      // 16384
    const int tilesPerWave = nTiles / wavesPerBatch;         // 32 (exact)

    for (int tIt = 0; tIt < tilesPerWave / 2; ++tIt) {       // 2 tiles / iter
        const int tileA = waveId * tilesPerWave + 2 * tIt;
        const int pA = tileA * 16 + m;       // tile A pixel for this lane
        const int pB = pA + 16;              // tile B pixel for this lane
        const float gxA = (float)(pA & (WD - 1));
        const float gyA = (float)(pA >> 9);
        const float gxB = (float)(pB & (WD - 1));
        const float gyB = (float)(pB >> 9);

        v2f bfA, bfB;
        bfA.x = lo ? gxA : fmaf(gxA, gxA, gyA * gyA);  // K0 / K2
        bfA.y = lo ? gyA : 1.0f;                       // K1 / K3
        bfB.x = lo ? gxB : fmaf(gxB, gxB, gyB * gyB);
        bfB.y = lo ? gyB : 1.0f;

        const v8f c0 = {};
        v8f d0A = __builtin_amdgcn_wmma_f32_16x16x4_f32(
            false, a0, false, bfA, (short)0, c0, false, false);
        v8f d1A = __builtin_amdgcn_wmma_f32_16x16x4_f32(
            false, a1, false, bfA, (short)0, c0, false, false);
        v8f d0B = __builtin_amdgcn_wmma_f32_16x16x4_f32(
            false, a0, false, bfB, (short)0, c0, false, false);
        v8f d1B = __builtin_amdgcn_wmma_f32_16x16x4_f32(
            false, a1, false, bfB, (short)0, c0, false, false);

        // t = r2 * log2(r2+eps); weights carry the 0.5*ln2 factor.
        // Raw v_log_f32: argument >= eps = 1e-7 -> never denormal/inf/NaN,
        // so the libm denorm/fixup sequence is unnecessary.
        float axA = 0.0f, ayA = 0.0f, axB = 0.0f, ayB = 0.0f;
        #pragma unroll
        for (int v = 0; v < 8; ++v) {
            const float tA0 = d0A[v] * __builtin_amdgcn_logf(d0A[v] + EPSF);
            axA = fmaf(tA0, wgx[v], axA);
            ayA = fmaf(tA0, wgy[v], ayA);
            const float tA1 = d1A[v] * __builtin_amdgcn_logf(d1A[v] + EPSF);
            axA = fmaf(tA1, wgx[8 + v], axA);
            ayA = fmaf(tA1, wgy[8 + v], ayA);
            const float tB0 = d0B[v] * __builtin_amdgcn_logf(d0B[v] + EPSF);
            axB = fmaf(tB0, wgx[v], axB);
            ayB = fmaf(tB0, wgy[v], ayB);
            const float tB1 = d1B[v] * __builtin_amdgcn_logf(d1B[v] + EPSF);
            axB = fmaf(tB1, wgx[8 + v], axB);
            ayB = fmaf(tB1, wgy[8 + v], ayB);
        }
        // other 8+8 sources live in the opposite lane half
        axA += __shfl_xor(axA, 16, 32);
        ayA += __shfl_xor(ayA, 16, 32);
        axB += __shfl_xor(axB, 16, 32);
        ayB += __shfl_xor(ayB, 16, 32);

        const float wxpA = axA + A00 + gxA * A10 + gyA * A20;
        const float wypA = ayA + A01 + gxA * A11 + gyA * A21;
        const float wxpB = axB + A00 + gxB * A10 + gyB * A20;
        const float wypB = ayB + A01 + gxB * A11 + gyB * A21;

        // Lane halves split the two tiles -> full EXEC through sampling.
        const float wxp = lo ? wxpA : wxpB;
        const float wyp = lo ? wypA : wypB;
        const int   p   = lo ? pA   : pB;
        const int   pw  = p & (WD - 1);
        const int   ph  = p >> 9;

        // normalize then un-normalize exactly as the reference does
        const float gx2 = -1.0f + wxp * (2.0f / (WD - 1));
        const float gy2 = -1.0f + wyp * (2.0f / (HD - 1));
        const float ix = ((gx2 + 1.0f) * WD - 1.0f) * 0.5f;
        const float iy = ((gy2 + 1.0f) * HD - 1.0f) * 0.5f;
        const float x0 = floorf(ix), y0 = floorf(iy);
        const float x1 = x0 + 1.0f,  y1 = y0 + 1.0f;
        const float wx1 = ix - x0, wx0 = 1.0f - wx1;
        const float wy1 = iy - y0, wy0 = 1.0f - wy1;

        const bool vx0 = (x0 >= 0.0f) && (x0 <= (float)(WD - 1));
        const bool vx1 = (x1 >= 0.0f) && (x1 <= (float)(WD - 1));
        const bool vy0 = (y0 >= 0.0f) && (y0 <= (float)(HD - 1));
        const bool vy1 = (y1 >= 0.0f) && (y1 <= (float)(HD - 1));
        const int xi0 = (int)fminf(fmaxf(x0, 0.0f), (float)(WD - 1));
        const int xi1 = (int)fminf(fmaxf(x1, 0.0f), (float)(WD - 1));
        const int yi0 = (int)fminf(fmaxf(y0, 0.0f), (float)(HD - 1));
        const int yi1 = (int)fminf(fmaxf(y1, 0.0f), (float)(HD - 1));

        const float w00 = (vx0 && vy0) ? wx0 * wy0 : 0.0f;
        const float w10 = (vx1 && vy0) ? wx1 * wy0 : 0.0f;
        const float w01 = (vx0 && vy1) ? wx0 * wy1 : 0.0f;
        const float w11 = (vx1 && vy1) ? wx1 * wy1 : 0.0f;

        #pragma unroll
        for (int c = 0; c < 3; ++c) {
            const float* ip = image + c * HD * WD;
            const float v00 = ip[yi0 * WD + xi0];
            const float v10 = ip[yi0 * WD + xi1];
            const float v01 = ip[yi1 * WD + xi0];
            const float v11 = ip[yi1 * WD + xi1];
            const float r = v00 * w00 + v10 * w10 + v01 * w01 + v11 * w11;
            out[(((size_t)b * 3 + c) * HD + ph) * WD + pw] = r;
        }
    }
}

// ---------------------------------------------------------------------------
extern "C" void kernel_launch(void* const* d_in, const int* in_sizes, int n_in,
                              void* d_out, int out_size, void* d_ws, size_t ws_size,
                              hipStream_t stream) {
    const float* x     = (const float*)d_in[0];   // (8,32,2)
    const float* poly  = (const float*)d_in[1];   // (1,32,2)
    const float* image = (const float*)d_in[2];   // (1,3,512,512)
    float* out         = (float*)d_out;           // (8,3,512,512)
    float* params      = (float*)d_ws;            // 8 * 35 * 2 floats

    tps_fit_kernel<<<8, 64, 0, stream>>>(x, poly, params);
    tps_warp_sample_kernel<<<8 * 64, 256, 0, stream>>>(x, image, params, out);
}